// RealAgnosticInteractionBlock_19653770347275
// MI455X (gfx1250) — compile-verified
//
#include <hip/hip_runtime.h>

#define N_MUL   128
#define NL      4
#define SUMD    16
#define NELEM   10
#define NNODES  8192
#define NEDGES  65536
#define EFEAT   8
#define HDIM    64
#define WNUMEL  512
#define H3STRIDE 65   // LDS row stride for TDM-padded h3 tile (64 DW + 1 DW pad)

typedef __attribute__((ext_vector_type(2))) float v2f;
typedef __attribute__((ext_vector_type(8))) float v8f;
typedef __attribute__((ext_vector_type(4))) unsigned int v4u;
typedef __attribute__((ext_vector_type(4))) int v4i;
typedef __attribute__((ext_vector_type(8))) int v8i;

__constant__ int c_LOF[SUMD] = {0,1,1,1,2,2,2,2,2,3,3,3,3,3,3,3};
__constant__ int c_OFFS[NL]  = {0,1,4,9};
__constant__ int c_DIMS[NL]  = {1,3,5,7};

// D = A(16x4,f32) * B(4x16,f32) + C(16x16,f32)  -- CDNA5 V_WMMA_F32_16X16X4_F32
__device__ __forceinline__ v8f wmma4(v2f a, v2f b, v8f c) {
  return __builtin_amdgcn_wmma_f32_16x16x4_f32(false, a, false, b, (short)0, c,
                                               false, false);
}

// ---------------------------------------------------------------------------
// x = node_feats @ W_up / sqrt(128).  grid=512 x 256 thr; wave = 16-col tile.
// ---------------------------------------------------------------------------
__global__ void k_node_up(const float* __restrict__ nf,
                          const float* __restrict__ Wup,
                          float* __restrict__ x) {
  const int lane = threadIdx.x & 31, wv = threadIdx.x >> 5;
  const int half = lane >> 4, lm = lane & 15, kb = half * 2;
  const int r0 = blockIdx.x * 16;
  const int c0 = wv * 16;
  v8f acc = {};
  for (int k0 = 0; k0 < N_MUL; k0 += 4) {
    v2f a, b;
    a.x = nf[(r0 + lm) * N_MUL + k0 + kb];
    a.y = nf[(r0 + lm) * N_MUL + k0 + kb + 1];
    b.x = Wup[(k0 + kb) * N_MUL + c0 + lm];
    b.y = Wup[(k0 + kb + 1) * N_MUL + c0 + lm];
    acc = wmma4(a, b, acc);
  }
  const float s = 0.08838834764831845f;  // 1/sqrt(128)
#pragma unroll
  for (int r = 0; r < 8; ++r) {
    const int row = r + half * 8;
    x[(r0 + row) * N_MUL + c0 + lm] = acc[r] * s;
  }
}

// ---------------------------------------------------------------------------
// Comb[l][z] = W_lin[l] @ W_skip[l][:,z,:] * (lin_norm/8 * skip_norm)
// ---------------------------------------------------------------------------
__global__ void k_comb(const float* __restrict__ Wlin,
                       const float* __restrict__ Wskip,
                       float* __restrict__ comb) {
  const int pair = blockIdx.x;
  const int l = pair / NELEM, wz = pair % NELEM;
  const float* A  = Wlin + (size_t)l * N_MUL * N_MUL;            // [u][v]
  const float* Bs = Wskip + (size_t)l * N_MUL * NELEM * N_MUL;   // [v][w][u2]
  float* C = comb + (size_t)pair * N_MUL * N_MUL;                // [u][u2]
  const int lane = threadIdx.x & 31, wv = threadIdx.x >> 5;
  const int half = lane >> 4, lm = lane & 15, kb = half * 2;
  const int c0 = wv * 16;
  const float scale = 3.0881591581494567e-4f;  // 1/(8*sqrt(128*1280))
  for (int rt = 0; rt < 8; ++rt) {
    const int r0 = rt * 16;
    v8f acc = {};
    for (int k0 = 0; k0 < N_MUL; k0 += 4) {
      v2f a, b;
      a.x = A[(r0 + lm) * N_MUL + k0 + kb];
      a.y = A[(r0 + lm) * N_MUL + k0 + kb + 1];
      b.x = Bs[(size_t)(k0 + kb) * NELEM * N_MUL + wz * N_MUL + c0 + lm];
      b.y = Bs[(size_t)(k0 + kb + 1) * NELEM * N_MUL + wz * N_MUL + c0 + lm];
      acc = wmma4(a, b, acc);
    }
#pragma unroll
    for (int r = 0; r < 8; ++r) {
      const int row = r + half * 8;
      C[(r0 + row) * N_MUL + c0 + lm] = acc[r] * scale;
    }
  }
}

// ---------------------------------------------------------------------------
// z[n] = argmax |node_attrs[n]|, av[n] = its value (== 1.0 for one-hot)
// ---------------------------------------------------------------------------
__global__ void k_z(const float* __restrict__ attrs, int* __restrict__ z,
                    float* __restrict__ av) {
  const int n = blockIdx.x * blockDim.x + threadIdx.x;
  if (n >= NNODES) return;
  int best = 0;
  float bv = attrs[(size_t)n * NELEM];
  for (int w = 1; w < NELEM; ++w) {
    const float v = attrs[(size_t)n * NELEM + w];
    if (fabsf(v) > fabsf(bv)) { bv = v; best = w; }
  }
  z[n] = best;
  av[n] = bv;
}

__global__ void k_bucket(const int* __restrict__ z, int* __restrict__ cnt,
                         int* __restrict__ bucket) {
  const int n = blockIdx.x * blockDim.x + threadIdx.x;
  if (n >= NNODES) return;
  const int zz = z[n];
  const int p = atomicAdd(&cnt[zz], 1);
  bucket[zz * NNODES + p] = n;
}

// ---------------------------------------------------------------------------
// 3-layer SiLU MLP. block = 128 thr (4 waves x 16 edges); grid = 1024.
// ---------------------------------------------------------------------------
__global__ void k_mlp3(const float* __restrict__ ef,
                       const float* __restrict__ W1, const float* __restrict__ b1,
                       const float* __restrict__ W2, const float* __restrict__ b2,
                       const float* __restrict__ W3, const float* __restrict__ b3,
                       float* __restrict__ h3) {
  __shared__ float hs[2][4][16][65];
  const int lane = threadIdx.x & 31, wv = threadIdx.x >> 5;
  const int half = lane >> 4, lm = lane & 15, kb = half * 2;
  const int e0 = (blockIdx.x * 4 + wv) * 16;

  for (int ct = 0; ct < 4; ++ct) {        // layer 1 (K=8)
    v8f acc = {};
    for (int k0 = 0; k0 < EFEAT; k0 += 4) {
      v2f a, b;
      a.x = ef[(e0 + lm) * EFEAT + k0 + kb];
      a.y = ef[(e0 + lm) * EFEAT + k0 + kb + 1];
      b.x = W1[(k0 + kb) * HDIM + ct * 16 + lm];
      b.y = W1[(k0 + kb + 1) * HDIM + ct * 16 + lm];
      acc = wmma4(a, b, acc);
    }
#pragma unroll
    for (int r = 0; r < 8; ++r) {
      const int row = r + half * 8, col = ct * 16 + lm;
      float v = acc[r] + b1[col];
      v = v / (1.0f + expf(-v));
      hs[0][wv][row][col] = v;
    }
  }
  __syncthreads();
  for (int ct = 0; ct < 4; ++ct) {        // layer 2 (K=64)
    v8f acc = {};
    for (int k0 = 0; k0 < HDIM; k0 += 4) {
      v2f a, b;
      a.x = hs[0][wv][lm][k0 + kb];
      a.y = hs[0][wv][lm][k0 + kb + 1];
      b.x = W2[(k0 + kb) * HDIM + ct * 16 + lm];
      b.y = W2[(k0 + kb + 1) * HDIM + ct * 16 + lm];
      acc = wmma4(a, b, acc);
    }
#pragma unroll
    for (int r = 0; r < 8; ++r) {
      const int row = r + half * 8, col = ct * 16 + lm;
      float v = acc[r] + b2[col];
      v = v / (1.0f + expf(-v));
      hs[1][wv][row][col] = v;
    }
  }
  __syncthreads();
  for (int ct = 0; ct < 4; ++ct) {        // layer 3 (K=64) -> global
    v8f acc = {};
    for (int k0 = 0; k0 < HDIM; k0 += 4) {
      v2f a, b;
      a.x = hs[1][wv][lm][k0 + kb];
      a.y = hs[1][wv][lm][k0 + kb + 1];
      b.x = W3[(k0 + kb) * HDIM + ct * 16 + lm];
      b.y = W3[(k0 + kb + 1) * HDIM + ct * 16 + lm];
      acc = wmma4(a, b, acc);
    }
#pragma unroll
    for (int r = 0; r < 8; ++r) {
      const int row = r + half * 8, col = ct * 16 + lm;
      float v = acc[r] + b3[col];
      v = v / (1.0f + expf(-v));
      h3[(size_t)(e0 + row) * HDIM + col] = v;
    }
  }
}

// ---------------------------------------------------------------------------
// k_msg: TDM-stage h3 tile (16x64 f32, LDS rows padded to 65 DW) once per
// block; tpw = h3 @ W4 + b4 via WMMA; then scatter
//   msgs[dst][mm][u] += tpw[e][l(mm)*128+u] * x[src][u] * Y[e][mm]
// block = 256 thr (8 waves), 16 edges/block; grid = 4096.
// ---------------------------------------------------------------------------
__global__ void k_msg(const float* __restrict__ h3, const float* __restrict__ W4,
                      const float* __restrict__ b4, const float* __restrict__ x,
                      const int* __restrict__ eidx,
                      const float* __restrict__ eattr,
                      float* __restrict__ msgs) {
  __shared__ float h3s[16][H3STRIDE];
  __shared__ float tpw[16][WNUMEL];
  const int lane = threadIdx.x & 31, wv = threadIdx.x >> 5;
  const int half = lane >> 4, lm = lane & 15, kb = half * 2;
  const int e0 = blockIdx.x * 16;

  if (wv == 0) {
    // Tensor DMA: 2D tile, 64 f32/row x 16 rows, dim0 stride 64, LDS padding
    // of 1 DWORD every 64 DWORDs -> LDS row stride 65 (bank-conflict-free).
    const unsigned long long ga =
        (unsigned long long)(uintptr_t)(h3 + (size_t)e0 * HDIM);
    const unsigned int ldsa = (unsigned int)(uintptr_t)(&h3s[0][0]);
    v4u g0;
    g0[0] = 1u;                                   // count=1, user descriptor
    g0[1] = ldsa;                                 // lds_addr
    g0[2] = (unsigned int)ga;                     // global_addr[31:0]
    g0[3] = ((unsigned int)(ga >> 32) & 0x01FFFFFFu) | (2u << 30); // +type=2
    v8i g1;
    g1[0] = (2 << 16)        // data_size = 4B
          | (1 << 20)        // pad_enable
          | (5 << 22)        // pad_interval: 64 DWORDs
          | (0 << 25);       // pad_amount: 1 DWORD
    g1[1] = (64 << 16);                 // tensor_dim0 = 64 (lo16)
    g1[2] = (16 << 16);                 // tensor_dim0 hi16=0 | tensor_dim1 = 16
    g1[3] = (64 << 16);                 // tensor_dim1 hi16=0 | tile_dim0 = 64
    g1[4] = 16;                         // tile_dim1 = 16, tile_dim2 = 0
    g1[5] = 64;                         // tensor_dim0_stride = 64
    g1[6] = 0;
    g1[7] = 0;
    v4i gz4 = {};
    v8i gz8 = {};
    __builtin_amdgcn_tensor_load_to_lds(g0, g1, gz4, gz4, gz8, 0);
    __builtin_amdgcn_s_wait_tensorcnt(0);
  }
  __syncthreads();

  for (int ct = 0; ct < 4; ++ct) {
    const int c0 = wv * 64 + ct * 16;
    v8f acc = {};
    for (int k0 = 0; k0 < HDIM; k0 += 4) {
      v2f a, b;
      a.x = h3s[lm][k0 + kb];
      a.y = h3s[lm][k0 + kb + 1];
      b.x = W4[(k0 + kb) * WNUMEL + c0 + lm];
      b.y = W4[(k0 + kb + 1) * WNUMEL + c0 + lm];
      acc = wmma4(a, b, acc);
    }
#pragma unroll
    for (int r = 0; r < 8; ++r) {
      const int row = r + half * 8;
      tpw[row][c0 + lm] = acc[r] + b4[c0 + lm];
    }
  }
  __syncthreads();

  const int u  = threadIdx.x >> 1;   // 0..127
  const int mh = threadIdx.x & 1;    // which 8 of the 16 mm slots
  for (int e = 0; e < 16; ++e) {
    const int ge = e0 + e;
    const int src = eidx[ge];
    const int dst = eidx[NEDGES + ge];
    const float xs = x[(size_t)src * N_MUL + u];
    float* mrow = msgs + (size_t)dst * (N_MUL * SUMD);
    const float* Y = eattr + (size_t)ge * SUMD;
#pragma unroll
    for (int j = 0; j < 8; ++j) {
      const int mm = mh * 8 + j;
      const int l = c_LOF[mm];
      const float t = tpw[e][l * N_MUL + u] * xs;
      atomicAdd(&mrow[mm * N_MUL + u], t * Y[mm]);
    }
  }
}

// ---------------------------------------------------------------------------
// out[n][u2][off_l+m] = av[n] * sum_u msgs[n][off_l+m][u] * Comb[l][z[n]][u][u2]
// grid=(40,GY); block = 256 thr (8 waves); wave wv owns u2 rows [wv*16,+16).
// Per column tile, B (128x16) is staged once into LDS (contiguous 512B/col),
// removing the 8x redundant global reads of msgs.
// ---------------------------------------------------------------------------
__global__ void k_out(const float* __restrict__ comb,
                      const float* __restrict__ msgs,
                      const int* __restrict__ cnt, const int* __restrict__ bucket,
                      const float* __restrict__ av, float* __restrict__ out) {
  __shared__ float Bs[128][17];
  const int pair = blockIdx.x;
  const int l = pair / NELEM, wz = pair % NELEM;
  const int d = c_DIMS[l], off = c_OFFS[l];
  const int ncols = cnt[wz] * d;
  const int lane = threadIdx.x & 31, wv = threadIdx.x >> 5;
  const int half = lane >> 4, lm = lane & 15, kb = half * 2;
  const int r0 = wv * 16;
  const float* A = comb + (size_t)pair * N_MUL * N_MUL;  // [u][u2]
  const int sn = threadIdx.x >> 4;   // staging: column 0..15
  const int su = threadIdx.x & 15;   // staging: 8-float chunk of u

  for (int ct = blockIdx.y; ct * 16 < ncols; ct += gridDim.y) {
    // ---- stage B tile: Bs[u][n] = msgs[node_n][mm_n][u]
    {
      const int colg = ct * 16 + sn;
      if (colg < ncols) {
        const int nd = bucket[wz * NNODES + colg / d];
        const int mm = off + colg % d;
        const float* src = msgs + (size_t)nd * (N_MUL * SUMD) + mm * N_MUL +
                           su * 8;
#pragma unroll
        for (int j = 0; j < 8; ++j) Bs[su * 8 + j][sn] = src[j];
      } else {
#pragma unroll
        for (int j = 0; j < 8; ++j) Bs[su * 8 + j][sn] = 0.0f;
      }
    }
    __syncthreads();

    v8f acc = {};
    for (int k0 = 0; k0 < N_MUL; k0 += 4) {
      v2f a, b;
      a.x = A[(k0 + kb) * N_MUL + r0 + lm];  // A[M=u2][K=u] = comb[u][u2]
      a.y = A[(k0 + kb + 1) * N_MUL + r0 + lm];
      b.x = Bs[k0 + kb][lm];
      b.y = Bs[k0 + kb + 1][lm];
      acc = wmma4(a, b, acc);
    }

    const int ccol = ct * 16 + lm;
    if (ccol < ncols) {
      const int node = bucket[wz * NNODES + ccol / d];
      const int m = ccol % d;
      const float s = av[node];
      float* op = out + (size_t)node * (N_MUL * SUMD) + off + m;
#pragma unroll
      for (int r = 0; r < 8; ++r) {
        const int row = r0 + r + half * 8;
        op[row * SUMD] = acc[r] * s;
      }
    }
    __syncthreads();   // Bs reused next iteration
  }
}

// ---------------------------------------------------------------------------
extern "C" void kernel_launch(void* const* d_in, const int* in_sizes, int n_in,
                              void* d_out, int out_size, void* d_ws,
                              size_t ws_size, hipStream_t stream) {
  (void)in_sizes; (void)n_in; (void)out_size; (void)ws_size;
  const float* node_attrs = (const float*)d_in[0];
  const float* node_feats = (const float*)d_in[1];
  const float* edge_attrs = (const float*)d_in[2];
  const float* edge_feats = (const float*)d_in[3];
  const int*   edge_index = (const int*)d_in[4];
  const float* W_up = (const float*)d_in[5];
  const float* W1 = (const float*)d_in[6];
  const float* b1 = (const float*)d_in[7];
  const float* W2 = (const float*)d_in[8];
  const float* b2 = (const float*)d_in[9];
  const float* W3 = (const float*)d_in[10];
  const float* b3 = (const float*)d_in[11];
  const float* W4 = (const float*)d_in[12];
  const float* b4 = (const float*)d_in[13];
  const float* W_lin  = (const float*)d_in[14];
  const float* W_skip = (const float*)d_in[15];

  char* ws = (char*)d_ws;
  float* x    = (float*)ws; ws += (size_t)NNODES * N_MUL * 4;          // 4 MB
  float* h3   = (float*)ws; ws += (size_t)NEDGES * HDIM * 4;           // 16 MB
  float* msgs = (float*)ws; ws += (size_t)NNODES * N_MUL * SUMD * 4;   // 64 MB
  float* comb = (float*)ws; ws += (size_t)NL * NELEM * N_MUL * N_MUL * 4;
  int*   zidx = (int*)ws;   ws += (size_t)NNODES * 4;
  float* av   = (float*)ws; ws += (size_t)NNODES * 4;
  int*   cnt  = (int*)ws;   ws += 64;
  int*   bucket = (int*)ws; ws += (size_t)NELEM * NNODES * 4;          // 320 KB

  (void)hipMemsetAsync(msgs, 0, (size_t)NNODES * N_MUL * SUMD * 4, stream);
  (void)hipMemsetAsync(cnt, 0, NELEM * sizeof(int), stream);

  k_node_up<<<NNODES / 16, 256, 0, stream>>>(node_feats, W_up, x);
  k_comb<<<NL * NELEM, 256, 0, stream>>>(W_lin, W_skip, comb);
  k_z<<<NNODES / 256, 256, 0, stream>>>(node_attrs, zidx, av);
  k_bucket<<<NNODES / 256, 256, 0, stream>>>(zidx, cnt, bucket);
  k_mlp3<<<NEDGES / 64, 128, 0, stream>>>(edge_feats, W1, b1, W2, b2, W3, b3, h3);
  k_msg<<<NEDGES / 16, 256, 0, stream>>>(h3, W4, b4, x, edge_index, edge_attrs,
                                         msgs);
  k_out<<<dim3(NL * NELEM, 128), 256, 0, stream>>>(comb, msgs, cnt, bucket, av,
                                                   (float*)d_out);
}